// OSI2_55791625175664
// MI455X (gfx1250) — compile-verified
//
#include <hip/hip_runtime.h>
#include <hip/hip_bf16.h>

// ---------------------------------------------------------------------------
// OSI2: 100-iteration Gaussian message passing on a 4x256x256 torus.
// [global max reduce] -> [packed-state init] -> 100x [step kernel]
// State packed as float4 {mu, sigma, rou0, rou1} per pixel, double-buffered
// in d_ws; momentum v packed as float4. mu also streamed to d_out each step
// (last write is the final answer).
// ---------------------------------------------------------------------------

#define HH 256
#define WW 256
#define HWMASK 65535

constexpr float SQRT2F = 1.41421356237309504880f;
constexpr float NDCF   = 0.91893853320467274178f;   // 0.5*log(2*pi)
constexpr float GA     = 1.22474487139158904910f;   // sqrt(1.5), GH node (f32)
constexpr float MOM    = 0.7f;
constexpr float LRATE  = 0.01f;
constexpr int   ITERS  = 100;

// One pairwise-quadrature edge evaluation (9 points), matching the JAX math.
__device__ __forceinline__ void edge_eval(
    float ew, float rou, float mu1, float sg1, float mu2, float sg2,
    float& o_dmu1, float& o_dsg1, float& o_drou, float& o_dmu2, float& o_dsg2)
{
    constexpr float XN[3] = {-GA, 0.0f, GA};
    constexpr float WN[3] = {1.0f/6.0f, 2.0f/3.0f, 1.0f/6.0f};

    float q   = 1.0f - rou * rou;
    float rtq = sqrtf(q);
    float s   = sqrtf(1.0f + rou) / 2.0f;
    float r   = sqrtf(1.0f - rou) / 2.0f;
    float sp  = s + r;
    float sm  = s - r;
    float cadd = 2.0f * NDCF + logf(sg1 * sg2) + logf(q) * 0.5f;
    float a1  = sg1 * SQRT2F;
    float a2  = sg2 * SQRT2F;
    float irt = 1.0f / rtq;

    float acc1 = 0.0f, acc2 = 0.0f, accr = 0.0f, accs1 = 0.0f, accs2 = 0.0f;
#pragma unroll
    for (int p = 0; p < 9; ++p) {
        const int   i   = p / 3, j = p % 3;
        const float xi  = XN[i], xj = XN[j];
        const float xij = xi * xj;                    // XIJ
        const float x2a = xi * xi + xj * xj - 1.0f;   // XI2AXJ2
        const float x2m = xi * xi - xj * xj;          // XI2MXJ2
        const float wep = WN[i] * WN[j];              // WE

        float z1 = sp * xi + sm * xj;
        float z2 = sm * xi + sp * xj;
        float fe = ew * (z1 * a1 + mu1) * (z2 * a2 + mu2);
        fe = (fe + cadd) * wep;

        acc1  += fe * (z1 - rou * z2);
        acc2  += fe * (z2 - rou * z1);
        accr  += fe * (2.0f * xij - rou * x2a);
        accs1 += fe * (x2a + x2m * irt);
        accs2 += fe * (x2a - x2m * irt);
    }
    o_dmu1 = acc1 / (sg1 * q) * SQRT2F;
    o_dmu2 = acc2 / (sg2 * q) * SQRT2F;
    o_drou = accr / q;
    o_dsg1 = accs1 / sg1;
    o_dsg2 = accs2 / sg2;
}

// ---- global max over edge_weight (order-independent, exact) ---------------
__global__ void kmax_part(const float* __restrict__ x, int n, float* __restrict__ part)
{
    __shared__ float smem[256];
    float m = -3.402823466e38f;
    for (int i = blockIdx.x * blockDim.x + threadIdx.x; i < n; i += gridDim.x * blockDim.x)
        m = fmaxf(m, x[i]);
    smem[threadIdx.x] = m;
    __syncthreads();
    for (int s = 128; s > 0; s >>= 1) {
        if (threadIdx.x < s) smem[threadIdx.x] = fmaxf(smem[threadIdx.x], smem[threadIdx.x + s]);
        __syncthreads();
    }
    if (threadIdx.x == 0) part[blockIdx.x] = smem[0];
}

__global__ void kmax_fin(const float* __restrict__ part, int np, float* __restrict__ denom)
{
    __shared__ float smem[256];
    float m = -3.402823466e38f;
    for (int i = threadIdx.x; i < np; i += blockDim.x) m = fmaxf(m, part[i]);
    smem[threadIdx.x] = m;
    __syncthreads();
    for (int s = 128; s > 0; s >>= 1) {
        if (threadIdx.x < s) smem[threadIdx.x] = fmaxf(smem[threadIdx.x], smem[threadIdx.x + s]);
        __syncthreads();
    }
    if (threadIdx.x == 0) denom[0] = smem[0] * 1.01f + 1.0f;
}

// ---- packed state init -----------------------------------------------------
// S[i] = {mu, sigma, rou0, rou1};  V[i] = momentum
__global__ void kinit(const float* __restrict__ y, const float* __restrict__ ew,
                      const float* __restrict__ denom,
                      float4* __restrict__ S, float4* __restrict__ V, int n)
{
    int i = blockIdx.x * blockDim.x + threadIdx.x;
    if (i >= n) return;
    float d = denom[0];
    const float2 e = ((const float2*)ew)[i];
    float4 s;
    s.x = y[i];
    s.y = 1.0f;
    s.z = e.x / d;
    s.w = e.y / d;
    S[i] = s;
    V[i] = make_float4(0.0f, 0.0f, 0.0f, 0.0f);
}

// ---- one solver iteration --------------------------------------------------
__global__ void __launch_bounds__(256)
kstep(const float4* __restrict__ Ssrc, float4* __restrict__ Sdst,
      float4* __restrict__ V,
      const float* __restrict__ y, const float* __restrict__ ew,
      const float* __restrict__ un, float* __restrict__ outMu, int n)
{
    int idx = blockIdx.x * blockDim.x + threadIdx.x;
    if (idx >= n) return;

    const int base = idx & ~HWMASK;            // batch plane base
    const int hw   = idx & HWMASK;
    const int h    = hw >> 8;
    const int w    = hw & (WW - 1);
    const int iDn  = base + (((h + 1) & (HH - 1)) << 8) + w;
    const int iRt  = base + (h << 8) + ((w + 1) & (WW - 1));
    const int iUp  = base + (((h - 1) & (HH - 1)) << 8) + w;
    const int iLf  = base + (h << 8) + ((w - 1) & (WW - 1));

    // gfx1250 global_prefetch_b8: pull neighbor-row state lines early
    __builtin_prefetch(&Ssrc[iDn], 0, 3);
    __builtin_prefetch(&Ssrc[iUp], 0, 3);

    const float4 sc = Ssrc[idx];                 // own {mu, sg, r0, r1}
    const float4 sD = Ssrc[iDn];
    const float4 sR = Ssrc[iRt];
    const float4 sU = Ssrc[iUp];
    const float4 sL = Ssrc[iLf];
    const float4 vv = V[idx];
    const float2 e2 = ((const float2*)ew)[idx];  // {ew0, ew1}
    const float  ewU = ew[2 * iUp];              // up neighbor, dir 0
    const float  ewL = ew[2 * iLf + 1];          // left neighbor, dir 1
    const float  yv  = y[idx];
    const float2 u2  = ((const float2*)un)[idx]; // {uw1, uw2}

    const float mu = sc.x, sg = sc.y, r0 = sc.z, r1 = sc.w;

    // unary quadrature (3 points)
    constexpr float XN[3] = {-GA, 0.0f, GA};
    constexpr float WN[3] = {1.0f/6.0f, 2.0f/3.0f, 1.0f/6.0f};
    float cu = 3.0f * (NDCF + logf(sg));
    float dmu = 0.0f, dsg = 0.0f;
#pragma unroll
    for (int k = 0; k < 3; ++k) {
        float xp = XN[k] * sg * SQRT2F + mu;
        float fn = xp * (u2.x * xp + u2.y * yv);
        fn = (fn - cu) * WN[k];
        dmu += fn * XN[k];
        dsg += fn * (2.0f * XN[k] * XN[k] - 1.0f);
    }
    dmu = dmu / sg * SQRT2F;
    dsg = dsg / sg;

    // own outgoing edges (down, right)
    float e0m1, e0s1, e0r, e0m2, e0s2;
    edge_eval(e2.x, r0, mu, sg, sD.x, sD.y, e0m1, e0s1, e0r, e0m2, e0s2);
    float e1m1, e1s1, e1r, e1m2, e1s2;
    edge_eval(e2.y, r1, mu, sg, sR.x, sR.y, e1m1, e1s1, e1r, e1m2, e1s2);
    // incoming edges, recomputed at this pixel (from up, from left)
    float um1, us1, ur, um2, us2;
    edge_eval(ewU, sU.z, sU.x, sU.y, mu, sg, um1, us1, ur, um2, us2);
    float lm1, ls1, lr, lm2, ls2;
    edge_eval(ewL, sL.w, sL.x, sL.y, mu, sg, lm1, ls1, lr, lm2, ls2);

    dmu += (e0m1 + e1m1);
    dmu += um2 + lm2;
    dsg += (e0s1 + e1s1);
    dsg += us2 + ls2;

    float4 nv;
    nv.x = MOM * vv.x + LRATE * dmu;
    nv.y = MOM * vv.y + LRATE * dsg;
    nv.z = MOM * vv.z + LRATE * e0r;
    nv.w = MOM * vv.w + LRATE * e1r;
    V[idx] = nv;

    float4 ns;
    ns.x = fminf(fmaxf(mu + nv.x, 0.0f),   63.0f);
    ns.y = fminf(fmaxf(sg + nv.y, 0.001f), 50.0f);
    ns.z = fminf(fmaxf(r0 + nv.z, -0.99f), 0.99f);
    ns.w = fminf(fmaxf(r1 + nv.w, -0.99f), 0.99f);
    Sdst[idx]  = ns;
    outMu[idx] = ns.x;   // last iteration's write is the final output
}

// ---------------------------------------------------------------------------
extern "C" void kernel_launch(void* const* d_in, const int* in_sizes, int n_in,
                              void* d_out, int out_size, void* d_ws, size_t ws_size,
                              hipStream_t stream)
{
    const float* y  = (const float*)d_in[0];   // [B,H,W]
    const float* ew = (const float*)d_in[1];   // [B,H,W,2]
    const float* un = (const float*)d_in[2];   // [B,H,W,2]
    float* out = (float*)d_out;
    const int n = in_sizes[0];                 // B*H*W (4*256*256)
    const size_t N = (size_t)n;

    float4* SA = (float4*)d_ws;                // N float4
    float4* SB = SA + N;                       // N float4
    float4* V  = SB + N;                       // N float4
    float* part  = (float*)(V + N);            // 256 floats
    float* denom = part + 256;                 // 1 float

    const int threads = 256;
    const int blocks  = (n + threads - 1) / threads;

    kmax_part<<<256, threads, 0, stream>>>(ew, 2 * n, part);
    kmax_fin<<<1, threads, 0, stream>>>(part, 256, denom);
    kinit<<<blocks, threads, 0, stream>>>(y, ew, denom, SA, V, n);

    float4* Ssrc = SA;
    float4* Sdst = SB;
    for (int it = 0; it < ITERS; ++it) {
        kstep<<<blocks, threads, 0, stream>>>(Ssrc, Sdst, V, y, ew, un, out, n);
        float4* t = Ssrc; Ssrc = Sdst; Sdst = t;
    }
}